// Analytic_net_86457691668663
// MI455X (gfx1250) — compile-verified
//
#include <hip/hip_runtime.h>

// Problem constants (from reference): E=1024, N=512, m=12, num_pts=4096, n=500
#define NN    512
#define EE    1024
#define MDIM  12
#define NPTS  4096
#define LD    544          // row stride of augmented matrix (f32 elems)
#define NAUG  528          // 512 + 16 augmented columns (12 RHS + 4 zero pad)
#define NBLK  8            // 512 / 64
#define BK    64           // LU block size

typedef float v2f __attribute__((ext_vector_type(2)));
typedef float v8f __attribute__((ext_vector_type(8)));

// ---------------------------------------------------------------------------
// Phase 1: A = constant_part + sum_e r[e] * M[e]   (1 GiB streaming read)
// 65536 threads, one float4 of the 512x512 result each, loop over e.
// ---------------------------------------------------------------------------
__global__ void k_reduce(const float* __restrict__ M, const float* __restrict__ r,
                         const float* __restrict__ cpart, float* __restrict__ Maug) {
  int idx = blockIdx.x * blockDim.x + threadIdx.x;   // 0..65535 float4 slots
  const float4* Mv = reinterpret_cast<const float4*>(M);
  const float4* Cv = reinterpret_cast<const float4*>(cpart);
  float4 s = Cv[idx];
  #pragma unroll 8
  for (int e = 0; e < EE; ++e) {
    float w = r[e];                                   // uniform -> s_load
    float4 mv = Mv[(size_t)e * 65536u + (size_t)idx];
    s.x += w * mv.x; s.y += w * mv.y; s.z += w * mv.z; s.w += w * mv.w;
  }
  int n = idx * 4;
  int row = n >> 9;          // /512
  int col = n & 511;
  *reinterpret_cast<float4*>(Maug + (size_t)row * LD + col) = s;
}

// Augmented columns 512..527: identity columns e_{500+j} (j<12), zeros beyond.
__global__ void k_initaug(float* __restrict__ Maug) {
  int idx = blockIdx.x * blockDim.x + threadIdx.x;   // 0..8191
  if (idx >= NN * 16) return;
  int row = idx >> 4;
  int j = idx & 15;
  Maug[(size_t)row * LD + 512 + j] = (j < MDIM && row == 500 + j) ? 1.0f : 0.0f;
}

// ---------------------------------------------------------------------------
// Panel LU with partial pivoting. One workgroup; tall panel (<=512x64 f32 =
// 128 KB) lives in dynamic LDS (CDNA5: 320 KB/WGP). Pivot rows recorded.
// ---------------------------------------------------------------------------
__global__ void k_panel(float* __restrict__ Maug, int* __restrict__ piv, int kb) {
  extern __shared__ char smem[];
  float* panel = reinterpret_cast<float*>(smem);              // 512*64 floats
  float* redv  = reinterpret_cast<float*>(smem + 512 * 64 * 4); // 256
  int*   redi  = reinterpret_cast<int*>(smem + 512 * 64 * 4 + 256 * 4); // 256
  int tid = threadIdx.x;
  int rb = kb * BK;
  int R = NN - rb;

  for (int idx = tid; idx < R * BK; idx += 256) {
    int rr = idx >> 6, cc = idx & 63;
    panel[idx] = Maug[(size_t)(rb + rr) * LD + rb + cc];
  }
  __syncthreads();

  for (int j = 0; j < BK; ++j) {
    // --- pivot search: argmax |panel[r][j]|, r in [j, R) ---
    float bv = -1.0f; int bi = j;
    for (int rr = j + tid; rr < R; rr += 256) {
      float v = fabsf(panel[rr * BK + j]);
      if (v > bv) { bv = v; bi = rr; }
    }
    redv[tid] = bv; redi[tid] = bi;
    __syncthreads();
    for (int s = 128; s > 0; s >>= 1) {
      if (tid < s) {
        float ov = redv[tid + s]; int oi = redi[tid + s];
        if (ov > redv[tid] || (ov == redv[tid] && oi < redi[tid])) {
          redv[tid] = ov; redi[tid] = oi;
        }
      }
      __syncthreads();
    }
    int p = redi[0];
    if (tid == 0) piv[rb + j] = rb + p;
    if (p != j && tid < BK) {
      float t = panel[j * BK + tid];
      panel[j * BK + tid] = panel[p * BK + tid];
      panel[p * BK + tid] = t;
    }
    __syncthreads();
    float inv = 1.0f / panel[j * BK + j];
    // scale column j below diag + rank-1 update of trailing panel
    for (int rr = j + 1 + tid; rr < R; rr += 256) {
      float l = panel[rr * BK + j] * inv;
      panel[rr * BK + j] = l;
      for (int cc = j + 1; cc < BK; ++cc)
        panel[rr * BK + cc] -= l * panel[j * BK + cc];
    }
    __syncthreads();
  }

  for (int idx = tid; idx < R * BK; idx += 256) {
    int rr = idx >> 6, cc = idx & 63;
    Maug[(size_t)(rb + rr) * LD + rb + cc] = panel[idx];
  }
}

// Apply this panel's 64 row swaps (as one permutation gather) to all columns
// outside the panel. 29 workgroups, 16-column slab each, slab in LDS.
__global__ void k_swap(float* __restrict__ Maug, const int* __restrict__ piv, int kb) {
  __shared__ float slab[NN * 16];   // 32 KB (rows rb..511 used)
  __shared__ int perm[NN];
  int tid = threadIdx.x;
  int rb = kb * BK;
  int R = NN - rb;
  int col0 = blockIdx.x * 16;
  if (col0 >= rb) col0 += BK;       // skip the panel's own columns

  if (tid == 0) {
    for (int i = 0; i < R; ++i) perm[i] = i;
    for (int j = 0; j < BK; ++j) {
      int p = piv[rb + j] - rb;
      int t = perm[j]; perm[j] = perm[p]; perm[p] = t;
    }
  }
  __syncthreads();
  for (int idx = tid; idx < R * 16; idx += 256) {
    int rr = idx >> 4, c = idx & 15;
    slab[idx] = Maug[(size_t)(rb + rr) * LD + col0 + c];
  }
  __syncthreads();
  for (int idx = tid; idx < R * 16; idx += 256) {
    int rr = idx >> 4, c = idx & 15;
    Maug[(size_t)(rb + rr) * LD + col0 + c] = slab[perm[rr] * 16 + c];
  }
}

// TRSM: U12 = L11^{-1} * A12 for the panel's block row (unit-lower L11),
// applied to all columns right of the panel incl. the augmented RHS columns.
__global__ void k_trsm(float* __restrict__ Maug, int kb, int ncols) {
  __shared__ float L11[BK * BK];    // 16 KB
  __shared__ float rhs[BK * BK];    // 16 KB
  int tid = threadIdx.x;
  int rb = kb * BK;
  int col0 = (kb + 1) * BK + blockIdx.x * BK;
  int cnt = ncols - blockIdx.x * BK; if (cnt > BK) cnt = BK;

  for (int idx = tid; idx < BK * BK; idx += 256)
    L11[idx] = Maug[(size_t)(rb + (idx >> 6)) * LD + rb + (idx & 63)];
  for (int idx = tid; idx < BK * BK; idx += 256) {
    int i = idx >> 6, c = idx & 63;
    rhs[idx] = (c < cnt) ? Maug[(size_t)(rb + i) * LD + col0 + c] : 0.0f;
  }
  __syncthreads();
  for (int j = 0; j < BK - 1; ++j) {
    for (int t = tid; t < (BK - 1 - j) * BK; t += 256) {
      int i = j + 1 + t / BK;
      int c = t & 63;
      rhs[i * BK + c] -= L11[i * BK + j] * rhs[j * BK + c];
    }
    __syncthreads();
  }
  for (int idx = tid; idx < BK * BK; idx += 256) {
    int i = idx >> 6, c = idx & 63;
    if (c < cnt) Maug[(size_t)(rb + i) * LD + col0 + c] = rhs[idx];
  }
}

// Trailing update A22 -= L21 @ U12 with v_wmma_f32_16x16x4_f32.
// One wave (32 lanes) per 16x16 output tile, K = 64 in 16 WMMA steps.
__global__ void k_gemm(float* __restrict__ Maug, int kb) {
  int lane = threadIdx.x;
  int half = lane >> 4;
  int lm = lane & 15;
  int rb2 = (kb + 1) * BK;
  int m0 = rb2 + blockIdx.y * 16;
  int n0 = rb2 + blockIdx.x * 16;
  int kc = kb * BK;
  v8f acc = {};
  #pragma unroll
  for (int kk = 0; kk < BK; kk += 4) {
    int ka = kc + kk + 2 * half;
    v2f a, b;
    a.x = Maug[(size_t)(m0 + lm) * LD + ka];
    a.y = Maug[(size_t)(m0 + lm) * LD + ka + 1];
    b.x = Maug[(size_t)ka * LD + n0 + lm];
    b.y = Maug[(size_t)(ka + 1) * LD + n0 + lm];
    acc = __builtin_amdgcn_wmma_f32_16x16x4_f32(false, a, false, b,
                                                (short)0, acc, false, false);
  }
  #pragma unroll
  for (int v = 0; v < 8; ++v) {
    int row = m0 + v + half * 8;
    Maug[(size_t)row * LD + n0 + lm] -= acc[v];
  }
}

// Blocked back-substitution: W = U^{-1} C for the 16 augmented columns.
// One workgroup; C (512x16) and the diagonal U block live in LDS.
__global__ void k_backsolve(const float* __restrict__ Maug, float* __restrict__ Wout) {
  __shared__ float cw[NN * 16];     // 32 KB
  __shared__ float ud[BK * BK];     // 16 KB
  int tid = threadIdx.x;

  for (int idx = tid; idx < NN * 16; idx += 256)
    cw[idx] = Maug[(size_t)(idx >> 4) * LD + 512 + (idx & 15)];
  __syncthreads();

  for (int kb = NBLK - 1; kb >= 0; --kb) {
    int rb = kb * BK;
    if (kb < NBLK - 1) {
      int c = tid & 15;
      for (int rr = tid >> 4; rr < BK; rr += 16) {
        int row = rb + rr;
        float acc = 0.0f;
        for (int j = rb + BK; j < NN; ++j)
          acc += Maug[(size_t)row * LD + j] * cw[j * 16 + c];
        cw[row * 16 + c] -= acc;
      }
    }
    __syncthreads();
    for (int idx = tid; idx < BK * BK; idx += 256)
      ud[idx] = Maug[(size_t)(rb + (idx >> 6)) * LD + rb + (idx & 63)];
    __syncthreads();
    for (int i = BK - 1; i >= 0; --i) {
      if (tid < 16) cw[(rb + i) * 16 + tid] /= ud[i * BK + i];
      __syncthreads();
      int c = tid & 15;
      for (int rr = tid >> 4; rr < i; rr += 16)
        cw[(rb + rr) * 16 + c] -= ud[rr * BK + i] * cw[(rb + i) * 16 + c];
      __syncthreads();
    }
  }
  for (int idx = tid; idx < NN * 16; idx += 256) Wout[idx] = cw[idx];
}

// Final GEMM: out(512x4096) = W(512x16) @ E(16x4096), E[j][p]=x[j*4096+p] (j<12).
// WMMA f32 16x16x4, one tile per wave, 8 waves per block. Memory bound (8 MiB).
__global__ void k_fgemm(const float* __restrict__ W, const float* __restrict__ x,
                        float* __restrict__ out) {
  int tid = threadIdx.x;
  int wid = tid >> 5;
  int lane = tid & 31;
  int half = lane >> 4;
  int lm = lane & 15;
  int tile = blockIdx.x * 8 + wid;            // 0..8191
  int tR = tile >> 8;                          // /256
  int tC = tile & 255;
  int m0 = tR * 16, n0 = tC * 16;
  v8f acc = {};
  #pragma unroll
  for (int kk = 0; kk < 16; kk += 4) {
    int ka = kk + 2 * half;
    v2f a, b;
    a.x = W[(m0 + lm) * 16 + ka];
    a.y = W[(m0 + lm) * 16 + ka + 1];
    b.x = (ka < MDIM) ? x[(size_t)ka * NPTS + n0 + lm] : 0.0f;
    b.y = (ka + 1 < MDIM) ? x[(size_t)(ka + 1) * NPTS + n0 + lm] : 0.0f;
    acc = __builtin_amdgcn_wmma_f32_16x16x4_f32(false, a, false, b,
                                                (short)0, acc, false, false);
  }
  #pragma unroll
  for (int v = 0; v < 8; ++v)
    out[(size_t)(m0 + v + half * 8) * NPTS + n0 + lm] = acc[v];
}

// ---------------------------------------------------------------------------
extern "C" void kernel_launch(void* const* d_in, const int* in_sizes, int n_in,
                              void* d_out, int out_size, void* d_ws, size_t ws_size,
                              hipStream_t stream) {
  const float* M     = (const float*)d_in[0];   // (1024,512,512)
  const float* r     = (const float*)d_in[1];   // (1024,)
  const float* cpart = (const float*)d_in[2];   // (512,512)
  const float* x     = (const float*)d_in[3];   // (4096,12) flat
  float* out = (float*)d_out;                   // (512,4096)

  char* ws = (char*)d_ws;
  float* Maug = (float*)ws;                               // 512*544 f32
  int*   piv  = (int*)(ws + (size_t)NN * LD * 4);         // 512 ints
  float* W    = (float*)(ws + (size_t)NN * LD * 4 + 2048);// 512*16 f32

  // Phase 1: build augmented [A | B]
  k_reduce<<<256, 256, 0, stream>>>(M, r, cpart, Maug);
  k_initaug<<<32, 256, 0, stream>>>(Maug);

  // Phase 2: blocked LU with partial pivoting (WMMA trailing updates)
  const size_t panel_lds = (size_t)512 * 64 * 4 + 256 * 4 + 256 * 4;
  for (int kb = 0; kb < NBLK; ++kb) {
    k_panel<<<1, 256, panel_lds, stream>>>(Maug, piv, kb);
    k_swap<<<(NAUG - BK) / 16, 256, 0, stream>>>(Maug, piv, kb);
    int ncols = NAUG - BK * (kb + 1);
    k_trsm<<<(ncols + BK - 1) / BK, 256, 0, stream>>>(Maug, kb, ncols);
    int rowsN = NN - BK * (kb + 1);
    if (rowsN > 0) {
      dim3 g(ncols / 16, rowsN / 16);
      k_gemm<<<g, 32, 0, stream>>>(Maug, kb);
    }
  }

  // Phase 3: back-substitution -> W = A^{-1}[:,500:512] (padded to 16 cols)
  k_backsolve<<<1, 256, 0, stream>>>(Maug, W);

  // Phase 4: out = W @ reshape(x, (12,4096))  (WMMA, memory bound)
  k_fgemm<<<1024, 256, 0, stream>>>(W, x, out);
}